// CategoricalDistInstance_18923625906267
// MI455X (gfx1250) — compile-verified
//
#include <hip/hip_runtime.h>

// Single-pass online softmax statistics, one 512-thread block per row.
// Data path: per-lane async global->LDS (ASYNCcnt) 8-deep pipeline,
// private LDS slots (no barriers in the hot loop), ds_load_b128 readback.
// HBM traffic = 524 MB read once -> ~22.5 us roofline at 23.3 TB/s.
// NBUF=8 keeps ~4KB in flight per wave so resident waves cover HBM latency.

#define L2E 1.4426950408889634f
#define LN2 0.6931471805599453f

constexpr int V      = 32000;
constexpr int NF4    = V / 4;              // 8000 float4 per row
constexpr int T      = 512;                // threads per block (16 waves)
constexpr int NBUF   = 8;                  // async pipeline depth
constexpr int NTILE  = (NF4 + T - 1) / T;  // 16 tiles
constexpr int MAINIT = NTILE - NBUF;       // 8 steady-state iterations

__device__ __forceinline__ float wexp2(float x) { return __builtin_amdgcn_exp2f(x); }
__device__ __forceinline__ float wlog2(float x) { return __builtin_amdgcn_logf(x); }

struct Acc { float m, s, t; }; // m: running max; s: sum exp2(a); t: sum a*exp2(a), a=(x-m)*L2E

// Rare path: full online update (new max seen in this float4).
__device__ __forceinline__ void upd_slow(float x, Acc &A) {
  float mn  = fmaxf(A.m, x);
  float a   = fmaxf((x   - mn) * L2E, -127.0f);
  float b   = fmaxf((A.m - mn) * L2E, -127.0f);
  float e   = wexp2(a);
  float c   = wexp2(b);
  float tmp = fmaf(b, A.s, A.t);
  A.t = fmaf(c, tmp, a * e);
  A.s = fmaf(c, A.s, e);
  A.m = mn;
}

// Hot path: max unchanged. 2 FMA + 1 add + 1 trans per element.
__device__ __forceinline__ void upd_fast(float x, float negm, Acc &A) {
  float a = fmaf(x, L2E, negm);
  float e = wexp2(a);          // underflows cleanly to 0
  A.s += e;
  A.t = fmaf(a, e, A.t);
}

__device__ __forceinline__ void proc4(float4 v, Acc &A, float &negm) {
  float mx = fmaxf(fmaxf(v.x, v.y), fmaxf(v.z, v.w));
  if (mx > A.m) {
    upd_slow(v.x, A); upd_slow(v.y, A); upd_slow(v.z, A); upd_slow(v.w, A);
    negm = -A.m * L2E;
  } else {
    upd_fast(v.x, negm, A); upd_fast(v.y, negm, A);
    upd_fast(v.z, negm, A); upd_fast(v.w, negm, A);
  }
}

__device__ __forceinline__ Acc merge(Acc a, Acc b) {
  float mm = fmaxf(a.m, b.m);
  float da = fmaxf((a.m - mm) * L2E, -127.0f);
  float db = fmaxf((b.m - mm) * L2E, -127.0f);
  float ca = wexp2(da);
  float cb = wexp2(db);
  Acc r;
  r.m = mm;
  r.s = fmaf(ca, a.s, cb * b.s);
  r.t = fmaf(ca, fmaf(da, a.s, a.t), cb * fmaf(db, b.s, b.t));
  return r;
}

__global__ __launch_bounds__(T)
void softmax_stats_kernel(const float* __restrict__ logits,
                          const int*   __restrict__ value,
                          float*       __restrict__ out, int B) {
  const int row = blockIdx.x;
  const int tid = threadIdx.x;
  const float* rowp = logits + (size_t)row * V;

  __shared__ float4 tile[NBUF][T];   // 64 KB staging, private slot per thread
  __shared__ float  red[3][T / 32];  // per-wave partials

  // LDS byte offset of this thread's buffer-0 slot (low 32 bits of flat LDS addr).
  unsigned lds0 = (unsigned)(size_t)(&tile[0][tid]);
  unsigned goff = (unsigned)(tid * 16);  // byte offset into row for tile 0

  // Prologue FIRST: start the async pipeline before anything that can stall.
  // Tiles 0..7 are always full (max index 7*512+511 = 4095 < 8000).
#pragma unroll
  for (int b = 0; b < NBUF; ++b) {
    unsigned lo = lds0 + (unsigned)(b * T * 16);
    unsigned go = goff + (unsigned)(b * T * 16);
    asm volatile("global_load_async_to_lds_b128 %0, %1, %2"
                 :: "v"(lo), "v"(go), "s"(rowp) : "memory");
  }

  // Dependent gather (consumed only by lane 0 at the very end). Issued after
  // the prologue so its scalar-load waits overlap with tile-0 async latency;
  // only wave 0 executes this block (EXECZ skip elsewhere).
  int   vi = 0;
  float xv = 0.0f;
  if (tid == 0) {
    vi = value[row];
    xv = rowp[vi];
  }

  Acc A; A.m = -3.0e38f; A.s = 0.0f; A.t = 0.0f;
  float negm = 0.0f;  // first proc4 always takes slow path and sets it

  // Steady state: tile indices tid + k*512 <= 4095 < 8000, no read guard needed.
  unsigned goff2 = goff + (unsigned)(NBUF * T * 16);
#pragma unroll
  for (int k = 0; k < MAINIT; ++k) {
    asm volatile("s_wait_asynccnt 7" ::: "memory");   // tile k landed in LDS
    float4 v = tile[k & (NBUF - 1)][tid];             // ds_load_b128
    asm volatile("s_wait_dscnt 0" ::: "memory");      // slot read complete before overwrite
    int idx2 = tid + (k + NBUF) * T;
    if (idx2 < NF4) {
      unsigned lo = lds0 + (unsigned)((k & (NBUF - 1)) * T * 16);
      asm volatile("global_load_async_to_lds_b128 %0, %1, %2"
                   :: "v"(lo), "v"(goff2), "s"(rowp) : "memory");
    }
    goff2 += (unsigned)(T * 16);
    proc4(v, A, negm);
  }

  // Epilogue: drain the pipeline fully, then process remaining tiles (guarded;
  // only tile 15 is partial: lanes tid < 320).
  asm volatile("s_wait_asynccnt 0" ::: "memory");
#pragma unroll
  for (int k = MAINIT; k < NTILE; ++k) {
    int idx = tid + k * T;
    if (idx < NF4) {
      float4 v = tile[k & (NBUF - 1)][tid];
      proc4(v, A, negm);
    }
  }

  // Intra-wave reduction (wave32).
#pragma unroll
  for (int d = 16; d >= 1; d >>= 1) {
    Acc Bp;
    Bp.m = __shfl_xor(A.m, d, 32);
    Bp.s = __shfl_xor(A.s, d, 32);
    Bp.t = __shfl_xor(A.t, d, 32);
    A = merge(A, Bp);
  }
  const int wave = tid >> 5, lane = tid & 31;
  if (lane == 0) { red[0][wave] = A.m; red[1][wave] = A.s; red[2][wave] = A.t; }
  __syncthreads();

  if (wave == 0) {
    Acc P; P.m = -3.0e38f; P.s = 0.0f; P.t = 0.0f;
    if (lane < T / 32) { P.m = red[0][lane]; P.s = red[1][lane]; P.t = red[2][lane]; }
#pragma unroll
    for (int d = (T / 32) / 2; d >= 1; d >>= 1) {
      Acc Bp;
      Bp.m = __shfl_xor(P.m, d, 32);
      Bp.s = __shfl_xor(P.s, d, 32);
      Bp.t = __shfl_xor(P.t, d, 32);
      P = merge(P, Bp);
    }
    if (lane == 0) {
      float l2s = wlog2(P.s);                 // log2(sum)
      float av  = (xv - P.m) * L2E;
      float lp2 = av - l2s;                   // log2(pdf)
      out[row]         = wexp2(lp2);          // pdf
      out[B + row]     = lp2 * LN2;           // log_prob (natural log)
      out[2 * B + row] = (P.t / P.s - l2s) * LN2;  // sum p*ln p (negative entropy)
    }
  }
}

extern "C" void kernel_launch(void* const* d_in, const int* in_sizes, int n_in,
                              void* d_out, int out_size, void* d_ws, size_t ws_size,
                              hipStream_t stream) {
  (void)n_in; (void)out_size; (void)d_ws; (void)ws_size;
  const float* logits = (const float*)d_in[0];
  const int*   value  = (const int*)d_in[1];
  float*       out    = (float*)d_out;
  const int    B      = in_sizes[1];  // 4096 rows
  softmax_stats_kernel<<<B, T, 0, stream>>>(logits, value, out, B);
}